// HGate_56573309222987
// MI455X (gfx1250) — compile-verified
//
#include <hip/hip_runtime.h>

// CDNA5 / gfx1250, wave32. Fused 5-qubit Hadamard on a 2^24 complex64 state.
// Complex 32x32 gate U = M^(kron5) expanded to the real-interleaved 64x64 matrix
// Ubar = [[Ur,-Ui],[Ui,Ur]] per element, so the interleaved (re,im) memory layout
// maps DIRECTLY onto V_WMMA_F32_16X16X4_F32 A/B/D fragments: b64 load -> B frag,
// D acc pair -> b64 store, no lane shuffles. Memory-bound: 256MB @ 23.3TB/s ~11us.
// Streams are read-once/write-once (> 192MB L2) -> non-temporal hints.

typedef __attribute__((ext_vector_type(2))) float v2f;
typedef __attribute__((ext_vector_type(8))) float v8f;

#define N_TILES 32768u   // 2^19 groups / 16 columns per tile

// Scatter 19-bit group index into the non-target bit positions of the 24-bit
// amplitude index. Target bits (LSB-relative): 3, 8, 13, 18, 23.
__device__ __forceinline__ unsigned expand_group(unsigned g) {
    return (g & 7u)
         | (((g >> 3)  & 15u) << 4)
         | (((g >> 7)  & 15u) << 9)
         | (((g >> 11) & 15u) << 14)
         | (((g >> 15) & 15u) << 19);
}

// Group-local complex index k (bit0 -> qudit 20 ... bit4 -> qudit 0) -> offset.
__device__ __forceinline__ unsigned koffset(unsigned k) {
    return ((k & 1u) << 3) | (((k >> 1) & 1u) << 8) | (((k >> 2) & 1u) << 13)
         | (((k >> 3) & 1u) << 18) | (((k >> 4) & 1u) << 23);
}

__global__ __launch_bounds__(256) void hgate5_wmma(const v2f* __restrict__ xv,
                                                   const float2* __restrict__ Mg,
                                                   v2f* __restrict__ outv) {
    const unsigned lane  = threadIdx.x & 31u;
    const unsigned wave  = threadIdx.x >> 5;
    const unsigned wpb   = blockDim.x >> 5;
    const unsigned wid   = blockIdx.x * wpb + wave;
    const unsigned nwave = gridDim.x * wpb;
    const unsigned h     = lane >> 4;       // WMMA lane half
    const unsigned lrow  = lane & 15u;
    const unsigned rc    = lrow & 1u;       // A-row parity: 0 = re row, 1 = im row

    // 2x2 gate (complex64)
    float2 m[2][2];
    m[0][0] = Mg[0]; m[0][1] = Mg[1];
    m[1][0] = Mg[2]; m[1][1] = Mg[3];

    // A fragments of Ubar (64x64), f32 16x4 layout, per (mb 0..3, kc 0..15):
    //   lane row   = 16*mb + lrow  -> complex m = 8*mb + (lrow>>1), parity rc
    //   elem r col = 4*kc + 2*h + r -> complex k = 2*kc + h, parity r
    //   rc=0: {Ur, -Ui}   rc=1: {Ui, Ur}
    v2f A[4][16];
#pragma unroll
    for (int mb = 0; mb < 4; ++mb) {
        const unsigned mrow = 8u * (unsigned)mb + (lrow >> 1);
#pragma unroll
        for (int kc = 0; kc < 16; ++kc) {
            const unsigned kcol = 2u * (unsigned)kc + h;
            float ur = 1.0f, ui = 0.0f;     // U[mrow][kcol] = prod_j M[m_j][k_j]
#pragma unroll
            for (int j = 0; j < 5; ++j) {
                float2 f = m[(mrow >> j) & 1u][(kcol >> j) & 1u];
                float nr = ur * f.x - ui * f.y;
                float ni = ur * f.y + ui * f.x;
                ur = nr; ui = ni;
            }
            A[mb][kc][0] = rc ? ui : ur;
            A[mb][kc][1] = rc ? ur : -ui;
        }
    }

    for (unsigned tile = wid; tile < N_TILES; tile += nwave) {
        // This lane's column: group id = tile*16 + (lane&15); shared by B and D.
        const unsigned base = expand_group(tile * 16u + lrow);

        // B fragments: one complex64 (b64, non-temporal) load per K-chunk lands
        // directly in the 4x16 fragment layout.
        v2f B[16];
#pragma unroll
        for (int kc = 0; kc < 16; ++kc)
            B[kc] = __builtin_nontemporal_load(
                        xv + (base + koffset(2u * (unsigned)kc + h)));

        v8f acc[4] = {};
#pragma unroll
        for (int kc = 0; kc < 16; ++kc) {
#pragma unroll
            for (int mb = 0; mb < 4; ++mb) {
                acc[mb] = __builtin_amdgcn_wmma_f32_16x16x4_f32(
                    false, A[mb][kc], false, B[kc], (short)0, acc[mb],
                    false, false);
            }
        }

        // D rows alternate re/im: complex out m = 8*mb + 4*h + t sits in
        // accumulator elements {2t, 2t+1} (adjacent VGPRs) -> direct b64 store.
#pragma unroll
        for (int mb = 0; mb < 4; ++mb) {
#pragma unroll
            for (int t = 0; t < 4; ++t) {
                const unsigned mout = 8u * (unsigned)mb + 4u * h + (unsigned)t;
                v2f o;
                o[0] = acc[mb][2 * t];
                o[1] = acc[mb][2 * t + 1];
                __builtin_nontemporal_store(o, outv + (base + koffset(mout)));
            }
        }
    }
}

extern "C" void kernel_launch(void* const* d_in, const int* in_sizes, int n_in,
                              void* d_out, int out_size, void* d_ws, size_t ws_size,
                              hipStream_t stream) {
    (void)in_sizes; (void)n_in; (void)out_size; (void)d_ws; (void)ws_size;
    const v2f*    x  = (const v2f*)d_in[0];     // complex64 state, 2^24 elements
    const float2* Mg = (const float2*)d_in[1];  // complex64 2x2 gate
    v2f*          o  = (v2f*)d_out;             // complex64 output

    // 1024 blocks x 8 waves = 8192 waves; 32768 tiles -> 4 tiles/wave (amortizes
    // per-wave construction of the 128-VGPR Ubar fragment set).
    hipLaunchKernelGGL(hgate5_wmma, dim3(1024), dim3(256), 0, stream, x, Mg, o);
}